// HFEncoder_31877247271078
// MI455X (gfx1250) — compile-verified
//
#include <hip/hip_runtime.h>

#define B_ 16
#define L_ 2048
#define D_ 768
#define K_ 32

typedef __attribute__((ext_vector_type(2))) float v2f;
typedef __attribute__((ext_vector_type(8))) float v8f;

// ---------------- Pass 1: scores[b,l] = x[b,l,:].w + bias  (one wave per row)
__global__ __launch_bounds__(256) void scores_kernel(
    const float* __restrict__ x, const float* __restrict__ w,
    const float* __restrict__ bias, float* __restrict__ scores) {
  const int wid = threadIdx.x >> 5, lane = threadIdx.x & 31;
  const int row = blockIdx.x * 8 + wid;              // < B*L
  const float* xr = x + (size_t)row * D_;
  float acc = 0.f;
#pragma unroll
  for (int i = 0; i < D_ / 128; ++i) {
    const int j = i * 128 + lane * 4;
    const float4 xv = *(const float4*)(xr + j);
    const float4 wv = *(const float4*)(w + j);
    acc += xv.x * wv.x + xv.y * wv.y + xv.z * wv.z + xv.w * wv.w;
  }
#pragma unroll
  for (int off = 16; off >= 1; off >>= 1) acc += __shfl_xor(acc, off, 32);
  if (lane == 0) scores[row] = acc + bias[0];
}

// ---------------- Pass 2: per-span max, 1/sum(exp), sent flag (one wave per span)
__global__ __launch_bounds__(256) void span_stats_kernel(
    const float* __restrict__ scores, const float* __restrict__ amask,
    const int* __restrict__ starts, const int* __restrict__ ends,
    float* __restrict__ mx_out, float* __restrict__ inv_out,
    float* __restrict__ sent_out) {
  const int wid = threadIdx.x >> 5, lane = threadIdx.x & 31;
  const int span = blockIdx.x * 8 + wid;             // b*K + k
  const int b = span / K_;
  int s = starts[span], e = ends[span];
  s = max(s, 0); e = min(e, L_);
  const float* sc = scores + b * L_;
  const float* am = amask + b * L_;
  float m = -1e30f;
  for (int l = s + lane; l < e; l += 32)
    if (am[l] >= 0.5f) m = fmaxf(m, sc[l]);
#pragma unroll
  for (int off = 16; off >= 1; off >>= 1) m = fmaxf(m, __shfl_xor(m, off, 32));
  float sum = 0.f, cnt = 0.f;
  for (int l = s + lane; l < e; l += 32)
    if (am[l] >= 0.5f) { sum += expf(sc[l] - m); cnt = 1.f; }
#pragma unroll
  for (int off = 16; off >= 1; off >>= 1) {
    sum += __shfl_xor(sum, off, 32);
    cnt = fmaxf(cnt, __shfl_xor(cnt, off, 32));
  }
  if (lane == 0) {
    mx_out[span]   = m;
    inv_out[span]  = (sum > 0.f) ? (1.f / sum) : 0.f;
    sent_out[span] = cnt;
  }
}

// ---------------- Pass 3: H[b,k,:] = sum_l w[b,k,l] * x[b,l,:] via V_WMMA_F32_16X16X4_F32
// grid = (D/128, K/16, B); 8 waves/block, each wave owns a 16-wide d-tile.
__global__ __launch_bounds__(256) void span_pool_wmma_kernel(
    const float* __restrict__ x, const float* __restrict__ amask,
    const int* __restrict__ starts, const int* __restrict__ ends,
    const float* __restrict__ scores, const float* __restrict__ mx,
    const float* __restrict__ invs, float* __restrict__ H) {
  const int b   = blockIdx.z;
  const int m0  = blockIdx.y * 16;                   // first span row of this M-tile
  const int wid = threadIdx.x >> 5;
  const int lane = threadIdx.x & 31;
  const int n   = lane & 15;                         // column within 16-wide tile
  const int hh  = lane >> 4;                         // lane-half (0 / 1)
  const int d   = blockIdx.x * 128 + wid * 16 + n;   // global feature column

  __shared__ int   s_start[16], s_end[16], s_sid[64];
  __shared__ float s_mx[16], s_inv[16], s_w[64];

  if (threadIdx.x < 16) {
    const int k = b * K_ + m0 + threadIdx.x;
    s_start[threadIdx.x] = starts[k];
    s_end[threadIdx.x]   = ends[k];
    s_mx[threadIdx.x]    = mx[k];
    s_inv[threadIdx.x]   = invs[k];
  }
  __syncthreads();

  int lo = L_, hi = 0;
  for (int k = 0; k < 16; ++k) {
    if (s_start[k] < s_end[k]) {
      lo = min(lo, s_start[k]);
      hi = max(hi, s_end[k]);
    }
  }

  const float* xb  = x + (size_t)b * L_ * D_;
  const float* scb = scores + b * L_;
  const float* amb = amask + b * L_;

  v8f acc = {0.f, 0.f, 0.f, 0.f, 0.f, 0.f, 0.f, 0.f};

  for (int t0 = lo; t0 < hi; t0 += 64) {
    __syncthreads();   // protect s_w/s_sid from previous iteration's readers
    if (threadIdx.x < 64) {
      const int l = t0 + (int)threadIdx.x;
      float wgt = 0.f;
      int sid = -1;
      if (l < hi) {
        for (int k = 0; k < 16; ++k)
          if (l >= s_start[k] && l < s_end[k]) { sid = k; break; }
        if (sid >= 0 && amb[l] >= 0.5f)
          wgt = expf(scb[l] - s_mx[sid]) * s_inv[sid];
        else
          sid = -1;
      }
      s_w[threadIdx.x]   = wgt;
      s_sid[threadIdx.x] = sid;
      // speculative prefetch of the next token chunk's rows (global_prefetch_b8)
      const int lp = min(l + 64, L_ - 1);
      __builtin_prefetch(&xb[(size_t)lp * D_ + blockIdx.x * 128], 0, 1);
    }
    __syncthreads();

    const int tmax = min(64, hi - t0);               // uniform across block
    for (int tc = 0; tc < tmax; tc += 4) {
      // f32 A 16x4 layout: lanes 0-15 hold K=0,1 ; lanes 16-31 hold K=2,3
      const int i0   = tc + 2 * hh;
      const int sid0 = s_sid[i0], sid1 = s_sid[i0 + 1];
      const float w0 = s_w[i0],   w1 = s_w[i0 + 1];
      v2f a;
      a.x = (sid0 == n) ? w0 : 0.f;                  // row m == lane&15
      a.y = (sid1 == n) ? w1 : 0.f;
      // B 4x16: lane-half selects K pair, VGPR selects within pair
      const int lr0 = min(t0 + i0,     L_ - 1);
      const int lr1 = min(t0 + i0 + 1, L_ - 1);
      v2f bm;
      bm.x = xb[(size_t)lr0 * D_ + d];
      bm.y = xb[(size_t)lr1 * D_ + d];
      acc = __builtin_amdgcn_wmma_f32_16x16x4_f32(
          false, a, false, bm, (short)0, acc, false, false);
    }
  }

  // C/D 16x16 f32 layout: VGPR r -> row m0+r (lanes 0-15) / m0+r+8 (lanes 16-31)
#pragma unroll
  for (int r = 0; r < 8; ++r) {
    const int krow = m0 + r + 8 * hh;
    H[((size_t)(b * K_ + krow)) * D_ + d] = acc[r];
  }
}

extern "C" void kernel_launch(void* const* d_in, const int* in_sizes, int n_in,
                              void* d_out, int out_size, void* d_ws, size_t ws_size,
                              hipStream_t stream) {
  const float* x  = (const float*)d_in[0];   // [B,L,D]
  const float* am = (const float*)d_in[1];   // [B,L]
  const int*   st = (const int*)d_in[2];     // [B,K]
  const int*   en = (const int*)d_in[3];     // [B,K]
  const float* wp = (const float*)d_in[4];   // [D]
  const float* bp = (const float*)d_in[5];   // [1]

  float* H    = (float*)d_out;                       // [B,K,D]
  float* sent = H + (size_t)B_ * K_ * D_;            // [B,K]

  float* scores = (float*)d_ws;                      // B*L
  float* mx     = scores + (size_t)B_ * L_;          // B*K
  float* invs   = mx + (size_t)B_ * K_;              // B*K

  scores_kernel<<<dim3((B_ * L_) / 8), dim3(256), 0, stream>>>(x, wp, bp, scores);
  span_stats_kernel<<<dim3((B_ * K_) / 8), dim3(256), 0, stream>>>(
      scores, am, st, en, mx, invs, sent);
  span_pool_wmma_kernel<<<dim3(D_ / 128, K_ / 16, B_), dim3(256), 0, stream>>>(
      x, am, st, en, scores, mx, invs, H);
}